// PointSegBatchlossHead_56822417326458
// MI455X (gfx1250) — compile-verified
//
#include <hip/hip_runtime.h>
#include <hip/hip_bf16.h>

typedef float v2f __attribute__((ext_vector_type(2)));
typedef float v8f __attribute__((ext_vector_type(8)));

#define EPS_BN_CLS   1e-5f
#define EPS_BN_ALIGN 1e-6f
#define EPS_NN       1e-8f

__device__ __forceinline__ v8f wmma4(v2f a, v2f b, v8f c) {
  // D = A(16x4 f32) * B(4x16 f32) + C(16x16 f32)
  return __builtin_amdgcn_wmma_f32_16x16x4_f32(false, a, false, b, (short)0, c,
                                               false, false);
}

__device__ __forceinline__ v8f vzero8() {
  v8f z = {0.f, 0.f, 0.f, 0.f, 0.f, 0.f, 0.f, 0.f};
  return z;
}

// Async copy of 16 bytes global -> LDS (bypasses VGPRs, tracked by ASYNCcnt).
__device__ __forceinline__ void async_copy16(const void* g, void* l) {
  const unsigned lds = (unsigned)(size_t)l;               // low 32b = LDS offset
  const unsigned long long ga = (unsigned long long)(size_t)g;
  asm volatile("global_load_async_to_lds_b128 %0, %1, off"
               :: "v"(lds), "v"(ga) : "memory");
}

__device__ __forceinline__ void wait_async0() {
  asm volatile("s_wait_asynccnt 0x0" ::: "memory");
}

// ---------------------------------------------------------------------------
// Kernel A: conv_cls head.  h = relu(bn(X @ W1)); out = h @ W2 + b2
// Weights staged in LDS k-pair-interleaved so every B-frag is one ds_load_b64.
// ---------------------------------------------------------------------------
__global__ __launch_bounds__(256)
void conv_cls_kernel(const float* __restrict__ X, const float* __restrict__ W1,
                     const float* __restrict__ bg, const float* __restrict__ bbeta,
                     const float* __restrict__ bm, const float* __restrict__ bv,
                     const float* __restrict__ W2, const float* __restrict__ bias2,
                     float* __restrict__ out, int nrows) {
  __shared__ float W1p[64 * 64];     // packed [(k>>1)][n][k&1], 16 KB
  __shared__ float W2p[64 * 32];     // packed, cols padded 20->32, 8 KB
  __shared__ float hid[8][16 * 68];  // per-wave 16x64 staging, stride 68

  const int tid = threadIdx.x;
  for (int d = tid; d < 4096; d += 256) {
    const int pb = d & 1, n = (d >> 1) & 63, kk = d >> 7;
    W1p[d] = W1[(2 * kk + pb) * 64 + n];
  }
  for (int d = tid; d < 2048; d += 256) {
    const int pb = d & 1, n = (d >> 1) & 31, kk = d >> 6;
    W2p[d] = (n < 20) ? W2[(2 * kk + pb) * 20 + n] : 0.0f;
  }
  __syncthreads();

  const int wave = tid >> 5, lane = tid & 31;
  const int m = lane & 15, half = lane >> 4;
  const int ntiles = nrows >> 4;
  int tile = blockIdx.x * 8 + wave;
  const bool valid = tile < ntiles;
  if (!valid) tile = ntiles - 1;
  const int row0 = tile << 4;
  const float* xrow = X + (size_t)(row0 + m) * 64 + 2 * half;

  v8f a0 = vzero8(), a1 = vzero8(), a2 = vzero8(), a3 = vzero8();
#pragma unroll
  for (int k0 = 0; k0 < 64; k0 += 4) {
    const int kk = (k0 >> 1) + half;  // K-pair index for this lane half
    v2f a  = *(const v2f*)(xrow + k0);
    v2f b0 = *(const v2f*)&W1p[(kk * 64 + m) * 2];
    v2f b1 = *(const v2f*)&W1p[(kk * 64 + 16 + m) * 2];
    v2f b2 = *(const v2f*)&W1p[(kk * 64 + 32 + m) * 2];
    v2f b3 = *(const v2f*)&W1p[(kk * 64 + 48 + m) * 2];
    a0 = wmma4(a, b0, a0);
    a1 = wmma4(a, b1, a1);
    a2 = wmma4(a, b2, a2);
    a3 = wmma4(a, b3, a3);
  }

  float* hw = hid[wave];
#pragma unroll
  for (int t = 0; t < 4; ++t) {
    const int ch = t * 16 + m;
    const float sc = bg[ch] * rsqrtf(bv[ch] + EPS_BN_CLS);
    const float sh = bbeta[ch] - bm[ch] * sc;
    v8f acc = (t == 0) ? a0 : (t == 1) ? a1 : (t == 2) ? a2 : a3;
#pragma unroll
    for (int r = 0; r < 8; ++r)
      hw[(r + 8 * half) * 68 + ch] = fmaxf(acc[r] * sc + sh, 0.0f);
  }
  asm volatile("s_wait_dscnt 0" ::: "memory");  // wave-local LDS RAW fence

  v8f o0 = vzero8(), o1 = vzero8();
#pragma unroll
  for (int k0 = 0; k0 < 64; k0 += 4) {
    const int kr = k0 + 2 * half;
    const int kk = kr >> 1;
    v2f a  = *(const v2f*)&hw[m * 68 + kr];
    v2f c0 = *(const v2f*)&W2p[(kk * 32 + m) * 2];
    v2f c1 = *(const v2f*)&W2p[(kk * 32 + 16 + m) * 2];
    o0 = wmma4(a, c0, o0);
    o1 = wmma4(a, c1, o1);
  }

  if (valid) {
#pragma unroll
    for (int r = 0; r < 8; ++r) {
      const int row = row0 + r + 8 * half;
      out[row * 20 + m] = o0[r] + bias2[m];
      if (m + 16 < 20) out[row * 20 + 16 + m] = o1[r] + bias2[16 + m];
    }
  }
}

// ---------------------------------------------------------------------------
// Kernel B: brute-force 3-NN per query point.
// Candidate tiles double-buffered in LDS via GLOBAL_LOAD_ASYNC_TO_LDS_B128.
// grid = (ceil(npp/256), B); frame uniform per block.
// ---------------------------------------------------------------------------
__global__ __launch_bounds__(256)
void knn_kernel(const float* __restrict__ pts, const float* __restrict__ cpc,
                int npp, int ncp, int* __restrict__ oidx, float* __restrict__ ow) {
  __shared__ float4 tile[2][256];  // [bs,x,y,z] per candidate, 8 KB total
  const int frame = blockIdx.y;
  const int p = blockIdx.x * 256 + threadIdx.x;  // frame-local point id
  const bool act = p < npp;
  const int pc = act ? p : (npp - 1);
  const float* pr = pts + (size_t)(frame * npp + pc) * 4;
  const float px = pr[1], py = pr[2], pz = pr[3];

  float d0 = 3.4e38f, d1 = 3.4e38f, d2 = 3.4e38f;
  int i0 = 0, i1 = 0, i2 = 0;
  const int base = frame * ncp;
  const int ntile = (ncp + 255) >> 8;

  {  // prime buffer 0
    const int ci = threadIdx.x;
    const int cc = (ci < ncp) ? ci : 0;  // tail entries never read
    async_copy16(cpc + (size_t)(base + cc) * 4, &tile[0][threadIdx.x]);
  }

  int cur = 0;
  for (int t = 0; t < ntile; ++t) {
    wait_async0();        // my async copies for tile[cur] landed
    __syncthreads();      // everyone's copies landed
    if (t + 1 < ntile) {  // overlap next tile's fetch with this tile's scan
      const int ci = (t + 1) * 256 + threadIdx.x;
      const int cc = (ci < ncp) ? ci : 0;
      async_copy16(cpc + (size_t)(base + cc) * 4, &tile[cur ^ 1][threadIdx.x]);
    }

    int lim = ncp - (t << 8);
    if (lim > 256) lim = 256;
    const float4* tb = tile[cur];
#pragma unroll 4
    for (int j = 0; j < lim; ++j) {
      const float4 c = tb[j];  // one broadcast ds_load_b128
      const float dx = px - c.y, dy = py - c.z, dz = pz - c.w;
      const float d = fmaf(dx, dx, fmaf(dy, dy, dz * dz));
      const int cidx = base + (t << 8) + j;
      if (d < d2) {
        if (d < d1) {
          if (d < d0) { d2 = d1; i2 = i1; d1 = d0; i1 = i0; d0 = d; i0 = cidx; }
          else        { d2 = d1; i2 = i1; d1 = d;  i1 = cidx; }
        } else        { d2 = d;  i2 = cidx; }
      }
    }
    __syncthreads();  // done reading tile[cur] before it is re-issued
    cur ^= 1;
  }

  if (act) {
    const float w0 = 1.0f / (fmaxf(d0, 0.0f) + EPS_NN);
    const float w1 = 1.0f / (fmaxf(d1, 0.0f) + EPS_NN);
    const float w2 = 1.0f / (fmaxf(d2, 0.0f) + EPS_NN);
    const float inv = 1.0f / (w0 + w1 + w2);
    const int gp = frame * npp + p;
    oidx[gp * 3 + 0] = i0;  ow[gp * 3 + 0] = w0 * inv;
    oidx[gp * 3 + 1] = i1;  ow[gp * 3 + 1] = w1 * inv;
    oidx[gp * 3 + 2] = i2;  ow[gp * 3 + 2] = w2 * inv;
  }
}

// ---------------------------------------------------------------------------
// Kernel B2: coalesced weighted feature gather. 64 lanes per point.
// ---------------------------------------------------------------------------
__global__ __launch_bounds__(256)
void gather_kernel(const float* __restrict__ F, const int* __restrict__ idx,
                   const float* __restrict__ w, float* __restrict__ interp,
                   int nptot) {
  const int grp = threadIdx.x >> 6;
  const int c = threadIdx.x & 63;
  const int p = blockIdx.x * 4 + grp;
  if (p >= nptot) return;
  const int i0 = idx[p * 3 + 0], i1 = idx[p * 3 + 1], i2 = idx[p * 3 + 2];
  const float w0 = w[p * 3 + 0], w1 = w[p * 3 + 1], w2 = w[p * 3 + 2];
  interp[(size_t)p * 64 + c] = w0 * F[(size_t)i0 * 64 + c] +
                               w1 * F[(size_t)i1 * 64 + c] +
                               w2 * F[(size_t)i2 * 64 + c];
}

// ---------------------------------------------------------------------------
// Kernel C: align + out head.  f = relu(bn6(X@alW + alB));
// h2 = relu(bn5(f@oW1)); out = h2@oW2 + ob2.  Three chained WMMA GEMMs.
// One packed weight buffer, repacked al_w -> out_w1 between stages.
// ---------------------------------------------------------------------------
__global__ __launch_bounds__(256)
void out_head_kernel(const float* __restrict__ X,
                     const float* __restrict__ alW, const float* __restrict__ alB,
                     const float* __restrict__ ag, const float* __restrict__ abeta,
                     const float* __restrict__ am, const float* __restrict__ av,
                     const float* __restrict__ oW1,
                     const float* __restrict__ og, const float* __restrict__ obeta,
                     const float* __restrict__ om, const float* __restrict__ ov,
                     const float* __restrict__ oW2, const float* __restrict__ ob2,
                     float* __restrict__ out, int nrows) {
  __shared__ float Wp[64 * 64];      // packed weight (al_w, then out_w1), 16 KB
  __shared__ float W2p[64 * 32];     // packed out_w2, padded, 8 KB
  __shared__ float hid[8][16 * 68];  // staging (~35 KB)

  const int tid = threadIdx.x;
  for (int d = tid; d < 4096; d += 256) {
    const int pb = d & 1, n = (d >> 1) & 63, kk = d >> 7;
    Wp[d] = alW[(2 * kk + pb) * 64 + n];
  }
  for (int d = tid; d < 2048; d += 256) {
    const int pb = d & 1, n = (d >> 1) & 31, kk = d >> 6;
    W2p[d] = (n < 20) ? oW2[(2 * kk + pb) * 20 + n] : 0.0f;
  }
  __syncthreads();

  const int wave = tid >> 5, lane = tid & 31;
  const int m = lane & 15, half = lane >> 4;
  const int ntiles = nrows >> 4;
  int tile = blockIdx.x * 8 + wave;
  const bool valid = tile < ntiles;
  if (!valid) tile = ntiles - 1;
  const int row0 = tile << 4;
  const float* xrow = X + (size_t)(row0 + m) * 64 + 2 * half;
  float* hw = hid[wave];

  // ---- stage 1: X @ al_w, + bias, BN(1e-6), ReLU -> hid ----
  v8f a0 = vzero8(), a1 = vzero8(), a2 = vzero8(), a3 = vzero8();
#pragma unroll
  for (int k0 = 0; k0 < 64; k0 += 4) {
    const int kk = (k0 >> 1) + half;
    v2f a  = *(const v2f*)(xrow + k0);
    v2f b0 = *(const v2f*)&Wp[(kk * 64 + m) * 2];
    v2f b1 = *(const v2f*)&Wp[(kk * 64 + 16 + m) * 2];
    v2f b2 = *(const v2f*)&Wp[(kk * 64 + 32 + m) * 2];
    v2f b3 = *(const v2f*)&Wp[(kk * 64 + 48 + m) * 2];
    a0 = wmma4(a, b0, a0);
    a1 = wmma4(a, b1, a1);
    a2 = wmma4(a, b2, a2);
    a3 = wmma4(a, b3, a3);
  }
#pragma unroll
  for (int t = 0; t < 4; ++t) {
    const int ch = t * 16 + m;
    const float sc = ag[ch] * rsqrtf(av[ch] + EPS_BN_ALIGN);
    const float sh = abeta[ch] - am[ch] * sc;
    const float bs = alB[ch];
    v8f acc = (t == 0) ? a0 : (t == 1) ? a1 : (t == 2) ? a2 : a3;
#pragma unroll
    for (int r = 0; r < 8; ++r)
      hw[(r + 8 * half) * 68 + ch] = fmaxf((acc[r] + bs) * sc + sh, 0.0f);
  }

  // swap weight buffer: al_w -> out_w1 (uniform across the block)
  __syncthreads();
  for (int d = tid; d < 4096; d += 256) {
    const int pb = d & 1, n = (d >> 1) & 63, kk = d >> 7;
    Wp[d] = oW1[(2 * kk + pb) * 64 + n];
  }
  __syncthreads();

  // ---- stage 2: f @ out_w1, BN(1e-5), ReLU -> hid (DS in-order per wave) ----
  a0 = vzero8(); a1 = vzero8(); a2 = vzero8(); a3 = vzero8();
#pragma unroll
  for (int k0 = 0; k0 < 64; k0 += 4) {
    const int kr = k0 + 2 * half;
    const int kk = kr >> 1;
    v2f a  = *(const v2f*)&hw[m * 68 + kr];
    v2f b0 = *(const v2f*)&Wp[(kk * 64 + m) * 2];
    v2f b1 = *(const v2f*)&Wp[(kk * 64 + 16 + m) * 2];
    v2f b2 = *(const v2f*)&Wp[(kk * 64 + 32 + m) * 2];
    v2f b3 = *(const v2f*)&Wp[(kk * 64 + 48 + m) * 2];
    a0 = wmma4(a, b0, a0);
    a1 = wmma4(a, b1, a1);
    a2 = wmma4(a, b2, a2);
    a3 = wmma4(a, b3, a3);
  }
#pragma unroll
  for (int t = 0; t < 4; ++t) {
    const int ch = t * 16 + m;
    const float sc = og[ch] * rsqrtf(ov[ch] + EPS_BN_CLS);
    const float sh = obeta[ch] - om[ch] * sc;
    v8f acc = (t == 0) ? a0 : (t == 1) ? a1 : (t == 2) ? a2 : a3;
#pragma unroll
    for (int r = 0; r < 8; ++r)
      hw[(r + 8 * half) * 68 + ch] = fmaxf(acc[r] * sc + sh, 0.0f);
  }
  asm volatile("s_wait_dscnt 0" ::: "memory");

  // ---- stage 3: h2 @ out_w2 + b2 -> out ----
  v8f o0 = vzero8(), o1 = vzero8();
#pragma unroll
  for (int k0 = 0; k0 < 64; k0 += 4) {
    const int kr = k0 + 2 * half;
    const int kk = kr >> 1;
    v2f a  = *(const v2f*)&hw[m * 68 + kr];
    v2f c0 = *(const v2f*)&W2p[(kk * 32 + m) * 2];
    v2f c1 = *(const v2f*)&W2p[(kk * 32 + 16 + m) * 2];
    o0 = wmma4(a, c0, o0);
    o1 = wmma4(a, c1, o1);
  }
  if (valid) {
#pragma unroll
    for (int r = 0; r < 8; ++r) {
      const int row = row0 + r + 8 * half;
      out[row * 20 + m] = o0[r] + ob2[m];
      if (m + 16 < 20) out[row * 20 + 16 + m] = o1[r] + ob2[16 + m];
    }
  }
}

// ---------------------------------------------------------------------------
extern "C" void kernel_launch(void* const* d_in, const int* in_sizes, int n_in,
                              void* d_out, int out_size, void* d_ws, size_t ws_size,
                              hipStream_t stream) {
  const float* cpf   = (const float*)d_in[0];   // conv_point_features [B*NC,64]
  const float* cpc   = (const float*)d_in[1];   // conv_point_coords   [B*NC,4]
  const float* pts   = (const float*)d_in[2];   // points              [B*NP,4]
  const float* clsW1 = (const float*)d_in[3];
  const float* clsG  = (const float*)d_in[4];
  const float* clsB  = (const float*)d_in[5];
  const float* clsM  = (const float*)d_in[6];
  const float* clsV  = (const float*)d_in[7];
  const float* clsW2 = (const float*)d_in[8];
  const float* clsB2 = (const float*)d_in[9];
  const float* alW   = (const float*)d_in[10];
  const float* alB   = (const float*)d_in[11];
  const float* alG   = (const float*)d_in[12];
  const float* alBe  = (const float*)d_in[13];
  const float* alM   = (const float*)d_in[14];
  const float* alV   = (const float*)d_in[15];
  const float* oW1   = (const float*)d_in[16];
  const float* oG    = (const float*)d_in[17];
  const float* oBe   = (const float*)d_in[18];
  const float* oM    = (const float*)d_in[19];
  const float* oV    = (const float*)d_in[20];
  const float* oW2   = (const float*)d_in[21];
  const float* oB2   = (const float*)d_in[22];

  const int nc_rows = in_sizes[0] / 64;   // B*NC (20000)
  const int np_rows = in_sizes[2] / 4;    // B*NP (40000)
  const int B = 2;                        // reference module constant
  const int npp = np_rows / B;
  const int ncp = nc_rows / B;

  float* conv_logits = (float*)d_out;                         // [nc_rows,20]
  float* out_logits  = (float*)d_out + (size_t)nc_rows * 20;  // [np_rows,20]

  char* ws = (char*)d_ws;
  float* interp = (float*)ws;                                     // np_rows*64 f32
  int*   kidx   = (int*)(ws + (size_t)np_rows * 64 * 4);          // np_rows*3  i32
  float* kw     = (float*)(ws + (size_t)np_rows * 64 * 4
                              + (size_t)np_rows * 3 * 4);         // np_rows*3  f32

  // conv_cls head (independent of KNN path)
  const int ntA = nc_rows / 16;
  conv_cls_kernel<<<dim3((ntA + 7) / 8), 256, 0, stream>>>(
      cpf, clsW1, clsG, clsB, clsM, clsV, clsW2, clsB2, conv_logits, nc_rows);

  // 3-NN search + weights (async double-buffered candidate tiles)
  knn_kernel<<<dim3((npp + 255) / 256, B), 256, 0, stream>>>(
      pts, cpc, npp, ncp, kidx, kw);

  // weighted feature gather -> interp features
  gather_kernel<<<dim3((np_rows + 3) / 4), 256, 0, stream>>>(
      cpf, kidx, kw, interp, np_rows);

  // align + out head
  const int ntC = np_rows / 16;
  out_head_kernel<<<dim3((ntC + 7) / 8), 256, 0, stream>>>(
      interp, alW, alB, alG, alBe, alM, alV, oW1, oG, oBe, oM, oV, oW2, oB2,
      out_logits, np_rows);
}